// GATLayer_15779709845542
// MI455X (gfx1250) — compile-verified
//
#include <hip/hip_runtime.h>

typedef __attribute__((ext_vector_type(2))) float v2f;
typedef __attribute__((ext_vector_type(8))) float v8f;

#define IN_DIM    64
#define OUT_DIM   16
#define NUM_HEADS 4
#define HD        64          // NUM_HEADS * OUT_DIM
#define NEG_SLOPE 0.2f
#define LDS_STRIDE 65         // 64 + 1 pad -> conflict-free column reads

// ---------------------------------------------------------------------------
// Node kernel: feat = h@W_fc (WMMA), el/er = <feat, attn_{l,r}> per head,
//              out  = h@W_res (WMMA)  [initializes d_out].
// 1 block = 256 threads = 8 waves, handles 16 node rows.
//   waves 0..3: head w of feat (+ el/er),  waves 4..7: cols of resval.
// ---------------------------------------------------------------------------
__global__ __launch_bounds__(256) void gat_node_wmma(
    const float* __restrict__ h,    const float* __restrict__ Wfc,
    const float* __restrict__ Wres, const float* __restrict__ attnl,
    const float* __restrict__ attnr,
    float* __restrict__ feat, float* __restrict__ el, float* __restrict__ er,
    float* __restrict__ out, int N)
{
    __shared__ float lh[16 * LDS_STRIDE];
    const int rowbase = blockIdx.x * 16;

    // Cooperative stage of the 16x64 h tile into LDS (float4 per thread).
    {
        int idx = threadIdx.x * 4;          // 256 * 4 = 1024 floats
        int r = idx >> 6, c = idx & 63;
        int gr = rowbase + r;
        float4 v = {0.f, 0.f, 0.f, 0.f};
        if (gr < N) v = *(const float4*)(h + (size_t)gr * IN_DIM + c);
        lh[r * LDS_STRIDE + c + 0] = v.x;
        lh[r * LDS_STRIDE + c + 1] = v.y;
        lh[r * LDS_STRIDE + c + 2] = v.z;
        lh[r * LDS_STRIDE + c + 3] = v.w;
    }
    __syncthreads();

    const int wave  = threadIdx.x >> 5;
    const int lane  = threadIdx.x & 31;
    const int lrow  = lane & 15;            // M row (A) / N col (B,C,D)
    const int half  = lane >> 4;            // 0: K0/K1, rows 0..7; 1: K2/K3, rows 8..15
    const int isres = wave >> 2;
    const int head  = wave & 3;
    const int col   = head * OUT_DIM + lrow;   // output column within 64
    const float* W  = isres ? Wres : Wfc;

    // D = A(16x4) x B(4x16) + C, accumulated over K = 64 in steps of 4.
    v8f acc = {};
    #pragma unroll
    for (int kk = 0; kk < 16; ++kk) {
        const int k = kk * 4 + 2 * half;    // ISA layout: VGPR0=K0/K2, VGPR1=K1/K3
        v2f a, b;
        a.x = lh[lrow * LDS_STRIDE + k];
        a.y = lh[lrow * LDS_STRIDE + k + 1];
        b.x = W[(size_t)k       * HD + col];
        b.y = W[(size_t)(k + 1) * HD + col];
        acc = __builtin_amdgcn_wmma_f32_16x16x4_f32(
                  false, a, false, b, (short)0, acc, false, false);
    }

    if (!isres) {
        // feat tile for head `head`; D layout: acc[j] = D[j + 8*half][lrow]
        const float al = attnl[head * OUT_DIM + lrow];
        const float ar = attnr[head * OUT_DIM + lrow];
        #pragma unroll
        for (int j = 0; j < 8; ++j) {
            const int r = rowbase + j + 8 * half;
            const float v = acc[j];
            if (r < N) feat[(size_t)r * HD + col] = v;
            // per-row dot with attn vectors: reduce across the 16 lanes
            float vl = v * al, vr = v * ar;
            #pragma unroll
            for (int m = 8; m >= 1; m >>= 1) {
                vl += __shfl_xor(vl, m, 32);
                vr += __shfl_xor(vr, m, 32);
            }
            if (lrow == 0 && r < N) {
                el[(size_t)r * NUM_HEADS + head] = vl;
                er[(size_t)r * NUM_HEADS + head] = vr;
            }
        }
    } else {
        // residual tile -> initialize output
        #pragma unroll
        for (int j = 0; j < 8; ++j) {
            const int r = rowbase + j + 8 * half;
            if (r < N) out[(size_t)r * HD + head * OUT_DIM + lrow] = acc[j];
        }
    }
}

// ---------------------------------------------------------------------------
// Edge kernel: 16 lanes per edge. Each lane: softmax over the 4 heads
// (redundant, broadcast loads), then one float4 of feat[src] scaled by the
// head weight, scatter-added into out[dst] with f32 atomics.
// ---------------------------------------------------------------------------
__global__ __launch_bounds__(256) void gat_edge(
    const int* __restrict__ src, const int* __restrict__ dst,
    const float* __restrict__ feat, const float* __restrict__ el,
    const float* __restrict__ er, float* __restrict__ out, int E)
{
    const long long t = (long long)blockIdx.x * blockDim.x + threadIdx.x;
    const int e   = (int)(t >> 4);
    const int sub = (int)(t & 15);
    if (e >= E) return;

    const int s = src[e];
    const int d = dst[e];

    const float4 l4 = *(const float4*)(el + (size_t)s * NUM_HEADS);
    const float4 r4 = *(const float4*)(er + (size_t)d * NUM_HEADS);
    float e0 = l4.x + r4.x, e1 = l4.y + r4.y, e2 = l4.z + r4.z, e3 = l4.w + r4.w;
    e0 = e0 > 0.f ? e0 : NEG_SLOPE * e0;
    e1 = e1 > 0.f ? e1 : NEG_SLOPE * e1;
    e2 = e2 > 0.f ? e2 : NEG_SLOPE * e2;
    e3 = e3 > 0.f ? e3 : NEG_SLOPE * e3;
    const float mx = fmaxf(fmaxf(e0, e1), fmaxf(e2, e3));
    const float x0 = __expf(e0 - mx);
    const float x1 = __expf(e1 - mx);
    const float x2 = __expf(e2 - mx);
    const float x3 = __expf(e3 - mx);
    const float inv = 1.0f / (x0 + x1 + x2 + x3);

    const int hidx = sub >> 2;   // (sub*4)/16 -> head owning this float4
    float aw = (hidx == 0) ? x0 : (hidx == 1) ? x1 : (hidx == 2) ? x2 : x3;
    aw *= inv;

    const float4 f4 = *(const float4*)(feat + (size_t)s * HD + sub * 4);
    float* op = out + (size_t)d * HD + sub * 4;
    atomicAdd(op + 0, f4.x * aw);
    atomicAdd(op + 1, f4.y * aw);
    atomicAdd(op + 2, f4.z * aw);
    atomicAdd(op + 3, f4.w * aw);
}

// ---------------------------------------------------------------------------
extern "C" void kernel_launch(void* const* d_in, const int* in_sizes, int n_in,
                              void* d_out, int out_size, void* d_ws, size_t ws_size,
                              hipStream_t stream) {
    const float* h     = (const float*)d_in[0];
    const int*   src   = (const int*)  d_in[1];
    const int*   dst   = (const int*)  d_in[2];
    const float* Wfc   = (const float*)d_in[3];
    const float* attnl = (const float*)d_in[4];
    const float* attnr = (const float*)d_in[5];
    const float* Wres  = (const float*)d_in[6];

    const int N = in_sizes[0] / IN_DIM;
    const int E = in_sizes[1];

    float* out  = (float*)d_out;
    float* feat = (float*)d_ws;                       // N*64 floats
    float* el   = feat + (size_t)N * HD;              // N*4 floats
    float* er   = el   + (size_t)N * NUM_HEADS;       // N*4 floats

    const int nblocks = (N + 15) / 16;
    gat_node_wmma<<<nblocks, 256, 0, stream>>>(h, Wfc, Wres, attnl, attnr,
                                               feat, el, er, out, N);

    const long long tot = (long long)E * 16;
    const int eblocks = (int)((tot + 255) / 256);
    gat_edge<<<eblocks, 256, 0, stream>>>(src, dst, feat, el, er, out, E);
}